// SelectiveSSM_8899172237904
// MI455X (gfx1250) — compile-verified
//
#include <hip/hip_runtime.h>
#include <hip/hip_bf16.h>
#include <math.h>

typedef float v2f __attribute__((ext_vector_type(2)));
typedef float v8f __attribute__((ext_vector_type(8)));

#define B_SZ    4
#define LSEQ    4096
#define DMODEL  1024
#define DSTATE  16
#define DTRANK  64
#define KPROJ   (DTRANK + 2 * DSTATE)   // 96
#define MTOT    (B_SZ * LSEQ)           // 16384

// ---------------------------------------------------------------------------
// Kernel A: xp[m, n] = sum_k x[m, k] * W_x[n, k]
//   M = 16384, N = 96, K = 1024.
//   One wave32 per 16-row M-stripe; it computes the full 16x96 output row
//   (6 accumulators), so each x row is read from HBM exactly once.
//   A/B fragments loaded as exact global_load_b64 per lane:
//     lanes 0-15 hold K={k,k+1}, lanes 16-31 hold K={k+2,k+3}
//   via a per-lane base offset of 2*half elements (no divergent branches).
// ---------------------------------------------------------------------------
__global__ __launch_bounds__(256) void ssm_proj_kernel(
    const float* __restrict__ x, const float* __restrict__ Wx,
    float* __restrict__ xp) {
  const int lane  = threadIdx.x & 31;
  const int wave  = threadIdx.x >> 5;
  const int mTile = blockIdx.x * 8 + wave;      // 1024 stripes total
  const int l15   = lane & 15;
  const int half  = lane >> 4;

  const float* aPtr = x  + (size_t)(mTile * 16 + l15) * DMODEL + 2 * half;
  const float* bPtr = Wx + (size_t)l15 * DMODEL + 2 * half;

  v8f acc[6] = {};
  for (int k = 0; k < DMODEL; k += 4) {
    const v2f a = *(const v2f*)(aPtr + k);
#pragma unroll
    for (int n = 0; n < 6; ++n) {
      const v2f b = *(const v2f*)(bPtr + (size_t)n * 16 * DMODEL + k);
      acc[n] = __builtin_amdgcn_wmma_f32_16x16x4_f32(
          false, a, false, b, (short)0, acc[n], false, false);
    }
  }

  const int mBase = mTile * 16 + half * 8;
#pragma unroll
  for (int n = 0; n < 6; ++n) {
    const int col = n * 16 + l15;
#pragma unroll
    for (int r = 0; r < 8; ++r)
      xp[(size_t)(mBase + r) * KPROJ + col] = acc[n][r];
  }
}

// ---------------------------------------------------------------------------
// Kernel B: dt[m, d] = softplus( sum_r xp[m, r] * W_dt[d, r] + b_dt[d] )
//   M = 16384, N = 1024, K = 64.
//   One wave32 per (16-row M-stripe, 128-col N-group): 8 accumulators,
//   cutting xp re-reads 8x; W_dt (256 KB) broadcasts from L2.
// ---------------------------------------------------------------------------
__global__ __launch_bounds__(256) void ssm_dt_kernel(
    const float* __restrict__ xp, const float* __restrict__ Wdt,
    const float* __restrict__ bdt, float* __restrict__ dt) {
  const int lane  = threadIdx.x & 31;
  const int wave  = threadIdx.x >> 5;
  const int tile  = blockIdx.x * 8 + wave;      // 1024 * 8 = 8192 tiles
  const int mTile = tile >> 3;
  const int ng    = tile & 7;                   // N-group: 8 n-tiles each
  const int l15   = lane & 15;
  const int half  = lane >> 4;

  const float* aPtr = xp  + (size_t)(mTile * 16 + l15) * KPROJ + 2 * half;
  const float* bPtr = Wdt + (size_t)(ng * 128 + l15) * DTRANK + 2 * half;

  v8f acc[8] = {};
#pragma unroll
  for (int k = 0; k < DTRANK; k += 4) {
    const v2f a = *(const v2f*)(aPtr + k);
#pragma unroll
    for (int g = 0; g < 8; ++g) {
      const v2f b = *(const v2f*)(bPtr + (size_t)g * 16 * DTRANK + k);
      acc[g] = __builtin_amdgcn_wmma_f32_16x16x4_f32(
          false, a, false, b, (short)0, acc[g], false, false);
    }
  }

  const int mBase = mTile * 16 + half * 8;
#pragma unroll
  for (int g = 0; g < 8; ++g) {
    const int d = ng * 128 + g * 16 + l15;
    const float bias = bdt[d];
#pragma unroll
    for (int r = 0; r < 8; ++r) {
      const float v = acc[g][r] + bias;
      const float sp = (v > 20.0f) ? v : log1pf(__expf(v));
      dt[(size_t)(mBase + r) * DMODEL + d] = sp;
    }
  }
}

// ---------------------------------------------------------------------------
// Kernel C: sequential selective scan over L, fused output epilogue.
//   One thread per (b, d) channel; h[16] + A_row[16] in registers.
//   Shared B_t/C_t vectors staged through LDS in 8-step chunks.
// ---------------------------------------------------------------------------
__global__ __launch_bounds__(256) void ssm_scan_kernel(
    const float* __restrict__ x, const float* __restrict__ A_log,
    const float* __restrict__ Dpar, const float* __restrict__ xp,
    const float* __restrict__ dt, float* __restrict__ y) {
  __shared__ float bc[8][32];   // [step][0:16]=B_t, [16:32]=C_t

  const int tid = threadIdx.x;
  const int b = blockIdx.x >> 2;                    // 4 blocks per batch
  const int d = ((blockIdx.x & 3) << 8) + tid;      // channel index

  float Arow[DSTATE], h[DSTATE];
#pragma unroll
  for (int s = 0; s < DSTATE; ++s) {
    Arow[s] = -__expf(A_log[(size_t)d * DSTATE + s]);   // A = -exp(A_log)
    h[s] = 0.0f;
  }
  const float dval = Dpar[d];

  const size_t base_bl = (size_t)b * LSEQ;
  const int cstep = tid >> 5;
  const int cj = tid & 31;

  for (int l0 = 0; l0 < LSEQ; l0 += 8) {
    __syncthreads();
    bc[cstep][cj] = xp[(base_bl + l0 + cstep) * KPROJ + DTRANK + cj];
    __syncthreads();

#pragma unroll
    for (int ll = 0; ll < 8; ++ll) {
      const size_t idx = (base_bl + l0 + ll) * DMODEL + d;
      const float xv = x[idx];
      const float dtv = dt[idx];
      const float dbx = dtv * xv;
      float yv = 0.0f;
#pragma unroll
      for (int s = 0; s < DSTATE; ++s) {
        h[s] = __expf(dtv * Arow[s]) * h[s] + dbx * bc[ll][s];
        yv += h[s] * bc[ll][16 + s];
      }
      y[idx] = yv + xv * dval;
    }
  }
}

// ---------------------------------------------------------------------------
extern "C" void kernel_launch(void* const* d_in, const int* in_sizes, int n_in,
                              void* d_out, int out_size, void* d_ws, size_t ws_size,
                              hipStream_t stream) {
  (void)in_sizes; (void)n_in; (void)out_size; (void)ws_size;
  const float* x     = (const float*)d_in[0];
  const float* A_log = (const float*)d_in[1];
  const float* Dpar  = (const float*)d_in[2];
  const float* Wx    = (const float*)d_in[3];
  const float* Wdt   = (const float*)d_in[4];
  const float* bdt   = (const float*)d_in[5];
  float* y = (float*)d_out;

  float* xp = (float*)d_ws;                       // 16384 * 96  fp32 = 6.3 MB
  float* dt = xp + (size_t)MTOT * KPROJ;          // 16384 * 1024 fp32 = 67 MB

  // GEMM1: 1024 M-stripes (one wave each), 8 waves per block
  ssm_proj_kernel<<<(MTOT / 16) / 8, 256, 0, stream>>>(x, Wx, xp);
  // GEMM2 + softplus: 1024 M-stripes x 8 N-groups
  ssm_dt_kernel<<<(MTOT / 16) * 8 / 8, 256, 0, stream>>>(xp, Wdt, bdt, dt);
  // Scan: 4 batches x 4 channel-blocks of 256
  ssm_scan_kernel<<<B_SZ * (DMODEL / 256), 256, 0, stream>>>(x, A_log, Dpar, xp, dt, y);
}